// ActorCritic_28312424415246
// MI455X (gfx1250) — compile-verified
//
#include <hip/hip_runtime.h>
#include <hip/hip_bf16.h>

#define H1 256
#define OUTD 32
#define ACTION_DIM 64
#define NODE_DIM 128
#define POS_DIM 128
#define MAX_POS 5120
#define NEG_SLOPE 0.2f

typedef __attribute__((ext_vector_type(2))) float v2f;
typedef __attribute__((ext_vector_type(8))) float v8f;

// ---------------------------------------------------------------- helpers
__device__ __forceinline__ void atomicMaxF(float* addr, float val) {
    unsigned int* ua = (unsigned int*)addr;
    unsigned int old = *ua;
    while (__uint_as_float(old) < val) {
        unsigned int assumed = old;
        old = atomicCAS(ua, assumed, __float_as_uint(val));
        if (old == assumed) break;
    }
}

__device__ __forceinline__ float waveReduceSum(float v) {
    #pragma unroll
    for (int off = 16; off > 0; off >>= 1) v += __shfl_xor(v, off, 32);
    return v;
}

// ---------------------------------------------------------------- init
__global__ void fill_f32(float* p, float v, long n) {
    long i = (long)blockIdx.x * blockDim.x + threadIdx.x;
    long stride = (long)gridDim.x * blockDim.x;
    for (; i < n; i += stride) p[i] = v;
}

__global__ void zero_i32(int* p, int n) {
    int i = blockIdx.x * blockDim.x + threadIdx.x;
    if (i < n) p[i] = 0;
}

// ---------------------------------------------------------------- node feature build
// h[i] = concat(text_emb[x[i,0]] (64), type_emb[x[i,1]] (64))
__global__ void build_h(const int* x, const float* text_emb, const float* type_emb,
                        float* h, int N) {
    long idx = (long)blockIdx.x * blockDim.x + threadIdx.x;
    long total = (long)N * NODE_DIM;
    if (idx >= total) return;
    int i = (int)(idx >> 7);          // / 128
    int j = (int)(idx & 127);
    float v;
    if (j < 64) v = text_emb[(long)x[2 * i] * 64 + j];
    else        v = type_emb[(long)x[2 * i + 1] * 64 + (j - 64)];
    h[idx] = v;
}

// ---------------------------------------------------------------- constant pos vector
// posvec[j] = pos_table[MAX_POS,:] @ pos_W[:,j] + pos_b[j]   (reference quirk: pos==MAX_POS always)
__global__ void pos_proj(const float* pos_table, const float* pos_W, const float* pos_b,
                         float* posvec) {
    int j = threadIdx.x;  // 128 threads
    const float* row = pos_table + (long)MAX_POS * POS_DIM;
    float acc = pos_b[j];
    for (int k = 0; k < POS_DIM; ++k) acc += row[k] * pos_W[k * POS_DIM + j];
    posvec[j] = acc;
}

// ---------------------------------------------------------------- flow type histogram
__global__ void flow_count(const int* eattr, int* cnt, int E) {
    int e = blockIdx.x * blockDim.x + threadIdx.x;
    int stride = gridDim.x * blockDim.x;
    for (; e < E; e += stride) atomicAdd(&cnt[eattr[2 * e]], 1);
}

// ---------------------------------------------------------------- edge-attr projection table
// eaW[f][j] (f=0..2 flow rows, f=3 loop row) = (flow-row + posvec) @ We
template <int H>
__global__ void ea_proj(const float* flow_emb, const float* posvec, const int* cnt,
                        const float* We, float* eaW, float invE) {
    int f = blockIdx.x;   // 0..3
    int j = threadIdx.x;  // H threads
    float c0 = cnt[0] * invE, c1 = cnt[1] * invE, c2 = cnt[2] * invE;
    float acc = 0.f;
    for (int k = 0; k < 128; ++k) {
        float r;
        if (f < 3) r = flow_emb[f * 128 + k];
        else       r = c0 * flow_emb[k] + c1 * flow_emb[128 + k] + c2 * flow_emb[256 + k];
        r += posvec[k];
        acc += r * We[k * H + j];
    }
    eaW[f * H + j] = acc;
}

// ---------------------------------------------------------------- dual GEMM via V_WMMA_F32_16X16X4_F32
// O1 = A@W1 + b1 ; O2 = A@W2 + b2.  A:[M,K] row-major, W:[K,NC] row-major.
// Block = 256 threads (8 waves). blockIdx.x = row tile (16 rows). Wave w covers col
// tiles w, w+8, ... A tile is loaded once per K-step and reused for both WMMAs.
__global__ void gemm_dual_wmma(const float* __restrict__ A,
                               const float* __restrict__ W1, const float* __restrict__ b1,
                               float* __restrict__ O1,
                               const float* __restrict__ W2, const float* __restrict__ b2,
                               float* __restrict__ O2,
                               int K, int NC) {
    int wave = threadIdx.x >> 5;
    int lane = threadIdx.x & 31;
    int half = lane >> 4;     // 0: lanes 0-15, 1: lanes 16-31
    int r    = lane & 15;
    long rowBase = (long)blockIdx.x * 16;

    for (int ct = wave; ct * 16 < NC; ct += 8) {
        int colBase = ct * 16;
        v8f acc1 = {0.f, 0.f, 0.f, 0.f, 0.f, 0.f, 0.f, 0.f};
        v8f acc2 = {0.f, 0.f, 0.f, 0.f, 0.f, 0.f, 0.f, 0.f};
        for (int k0 = 0; k0 < K; k0 += 4) {
            int ka = k0 + 2 * half;  // A/B VGPR0 carries K = {k0, k0+2}, VGPR1 = {k0+1, k0+3}
            v2f a, bv1, bv2;
            const float* arow = A + (rowBase + r) * K + ka;
            a.x = arow[0];
            a.y = arow[1];
            bv1.x = W1[(long)ka * NC + colBase + r];
            bv1.y = W1[(long)(ka + 1) * NC + colBase + r];
            bv2.x = W2[(long)ka * NC + colBase + r];
            bv2.y = W2[(long)(ka + 1) * NC + colBase + r];
            acc1 = __builtin_amdgcn_wmma_f32_16x16x4_f32(false, a, false, bv1,
                                                         (short)0, acc1, false, false);
            acc2 = __builtin_amdgcn_wmma_f32_16x16x4_f32(false, a, false, bv2,
                                                         (short)0, acc2, false, false);
        }
        #pragma unroll
        for (int v = 0; v < 8; ++v) {
            long row = rowBase + v + 8 * half;   // C/D: VGPR v holds M=v (lanes<16), M=v+8
            int  col = colBase + r;
            O1[row * NC + col] = acc1[v] + b1[col];
            O2[row * NC + col] = acc2[v] + b2[col];
        }
    }
}

// ---------------------------------------------------------------- attention logits (edges)
// One wave per edge: e = leakyrelu(xl[s]+xr[d]+eaW[flow]) . att ; atomicMax into emax[d].
template <int H>
__global__ void edge_scores(const int* __restrict__ eidx, const int* __restrict__ eattr,
                            const float* __restrict__ xl, const float* __restrict__ xr,
                            const float* __restrict__ eaW, const float* __restrict__ att,
                            float* __restrict__ esc, float* __restrict__ emax, int E) {
    int e = blockIdx.x * (blockDim.x >> 5) + (threadIdx.x >> 5);
    if (e >= E) return;
    int lane = threadIdx.x & 31;
    int s = eidx[e];
    int d = eidx[E + e];
    int f = eattr[2 * e];
    float acc = 0.f;
    #pragma unroll
    for (int j = lane; j < H; j += 32) {
        float m = xl[(long)s * H + j] + xr[(long)d * H + j] + eaW[f * H + j];
        m = (m > 0.f) ? m : NEG_SLOPE * m;
        acc += m * att[j];
    }
    acc = waveReduceSum(acc);
    if (lane == 0) {
        esc[e] = acc;
        atomicMaxF(&emax[d], acc);
    }
}

// self-loop logits: s = d = i, ea row 3 (loop attr)
template <int H>
__global__ void self_scores(const float* __restrict__ xl, const float* __restrict__ xr,
                            const float* __restrict__ eaW, const float* __restrict__ att,
                            float* __restrict__ eself, float* __restrict__ emax, int N) {
    int i = blockIdx.x * (blockDim.x >> 5) + (threadIdx.x >> 5);
    if (i >= N) return;
    int lane = threadIdx.x & 31;
    float acc = 0.f;
    #pragma unroll
    for (int j = lane; j < H; j += 32) {
        float m = xl[(long)i * H + j] + xr[(long)i * H + j] + eaW[3 * H + j];
        m = (m > 0.f) ? m : NEG_SLOPE * m;
        acc += m * att[j];
    }
    acc = waveReduceSum(acc);
    if (lane == 0) {
        eself[i] = acc;
        atomicMaxF(&emax[i], acc);
    }
}

// ---------------------------------------------------------------- softmax numerator + denominator
// sc[t] <- exp(sc[t] - emax[dst]);  denom[dst] += sc[t].  dst==nullptr => dst=t (self loops).
__global__ void exp_accum(const int* dst, float* sc, const float* emax, float* denom, int n) {
    int t = blockIdx.x * blockDim.x + threadIdx.x;
    int stride = gridDim.x * blockDim.x;
    for (; t < n; t += stride) {
        int d = dst ? dst[t] : t;
        float a = expf(sc[t] - emax[d]);
        sc[t] = a;
        atomicAdd(&denom[d], a);
    }
}

// ---------------------------------------------------------------- output init: bias + self-loop term
template <int H>
__global__ void hout_init(const float* __restrict__ xl, const float* __restrict__ eself,
                          const float* __restrict__ denom, const float* __restrict__ bias,
                          float* __restrict__ hout, int N) {
    long idx = (long)blockIdx.x * blockDim.x + threadIdx.x;
    long total = (long)N * H;
    if (idx >= total) return;
    int i = (int)(idx / H);
    int j = (int)(idx % H);
    float alpha = eself[i] / denom[i];
    hout[idx] = bias[j] + alpha * xl[idx];
}

// ---------------------------------------------------------------- scatter: hout[d] += alpha * xl[s]
template <int H>
__global__ void edge_aggregate(const int* __restrict__ eidx, const float* __restrict__ esc,
                               const float* __restrict__ denom, const float* __restrict__ xl,
                               float* __restrict__ hout, int E) {
    int e = blockIdx.x * (blockDim.x >> 5) + (threadIdx.x >> 5);
    if (e >= E) return;
    int lane = threadIdx.x & 31;
    int s = eidx[e];
    int d = eidx[E + e];
    float alpha = esc[e] / denom[d];
    #pragma unroll
    for (int j = lane; j < H; j += 32)
        atomicAdd(&hout[(long)d * H + j], alpha * xl[(long)s * H + j]);
}

// ---------------------------------------------------------------- mean pool (OUTD cols)
__global__ void reduce_mean(const float* __restrict__ h2, float* __restrict__ gsum, int N) {
    int lane = threadIdx.x & 31;
    int wid  = (blockIdx.x * blockDim.x + threadIdx.x) >> 5;
    int nw   = (gridDim.x * blockDim.x) >> 5;
    float acc = 0.f;
    for (int i = wid; i < N; i += nw) acc += h2[(long)i * OUTD + lane];
    atomicAdd(&gsum[lane], acc);
}

// ---------------------------------------------------------------- policy head + softmax (1 block, 64 thr)
__global__ void policy_softmax(const float* __restrict__ gsum, const float* __restrict__ W,
                               const float* __restrict__ b, float* __restrict__ out, float invN) {
    __shared__ float red[ACTION_DIM];
    int a = threadIdx.x;  // 0..63
    float acc = b[a];
    for (int j = 0; j < OUTD; ++j) acc += gsum[j] * invN * W[j * ACTION_DIM + a];
    red[a] = acc;
    __syncthreads();
    float mx = red[0];
    for (int j = 1; j < ACTION_DIM; ++j) mx = fmaxf(mx, red[j]);
    float ex = expf(acc - mx);
    __syncthreads();
    red[a] = ex;
    __syncthreads();
    float sm = 0.f;
    for (int j = 0; j < ACTION_DIM; ++j) sm += red[j];
    out[a] = ex / sm;
}

// ================================================================ launcher
extern "C" void kernel_launch(void* const* d_in, const int* in_sizes, int n_in,
                              void* d_out, int out_size, void* d_ws, size_t ws_size,
                              hipStream_t stream) {
    const int N = in_sizes[0] / 2;
    const int E = in_sizes[1] / 2;

    const int*   x         = (const int*)d_in[0];
    const int*   eidx      = (const int*)d_in[1];
    const int*   eattr     = (const int*)d_in[2];
    const float* text_emb  = (const float*)d_in[3];
    const float* type_emb  = (const float*)d_in[4];
    const float* flow_emb  = (const float*)d_in[5];
    const float* pos_table = (const float*)d_in[6];
    const float* pos_W     = (const float*)d_in[7];
    const float* pos_b     = (const float*)d_in[8];
    const float* c1_Wl     = (const float*)d_in[9];
    const float* c1_bl     = (const float*)d_in[10];
    const float* c1_Wr     = (const float*)d_in[11];
    const float* c1_br     = (const float*)d_in[12];
    const float* c1_We     = (const float*)d_in[13];
    const float* c1_att    = (const float*)d_in[14];
    const float* c1_bias   = (const float*)d_in[15];
    const float* c2_Wl     = (const float*)d_in[16];
    const float* c2_bl     = (const float*)d_in[17];
    const float* c2_Wr     = (const float*)d_in[18];
    const float* c2_br     = (const float*)d_in[19];
    const float* c2_We     = (const float*)d_in[20];
    const float* c2_att    = (const float*)d_in[21];
    const float* c2_bias   = (const float*)d_in[22];
    const float* policy_W  = (const float*)d_in[23];
    const float* policy_b  = (const float*)d_in[24];

    // workspace carve (floats)
    float* ws = (float*)d_ws;
    size_t off = 0;
    float* h      = ws + off; off += (size_t)N * NODE_DIM;
    float* xl     = ws + off; off += (size_t)N * H1;     // layer2 reuses as xl2 (N*OUTD)
    float* xr     = ws + off; off += (size_t)N * H1;     // layer2 reuses as xr2
    float* h1     = ws + off; off += (size_t)N * H1;
    float* h2     = ws + off; off += (size_t)N * OUTD;
    float* esc    = ws + off; off += (size_t)E;
    float* eself  = ws + off; off += (size_t)N;
    float* emax   = ws + off; off += (size_t)N;
    float* denom  = ws + off; off += (size_t)N;
    float* posvec = ws + off; off += POS_DIM;
    float* eaW1   = ws + off; off += 4 * H1;
    float* eaW2   = ws + off; off += 4 * OUTD;
    float* gsum   = ws + off; off += ACTION_DIM;
    int*   cnt    = (int*)(ws + off); off += 16;

    const int wavesPerBlock = 8;      // 256 threads
    const int eBlocks = (E + wavesPerBlock - 1) / wavesPerBlock;
    const int nBlocks = (N + wavesPerBlock - 1) / wavesPerBlock;

    // --- preprocessing ----------------------------------------------------
    {
        long tot = (long)N * NODE_DIM;
        build_h<<<(int)((tot + 255) / 256), 256, 0, stream>>>(x, text_emb, type_emb, h, N);
    }
    zero_i32<<<1, 16, 0, stream>>>(cnt, 16);
    flow_count<<<512, 256, 0, stream>>>(eattr, cnt, E);
    pos_proj<<<1, POS_DIM, 0, stream>>>(pos_table, pos_W, pos_b, posvec);
    ea_proj<H1><<<4, H1, 0, stream>>>(flow_emb, posvec, cnt, c1_We, eaW1, 1.f / (float)E);
    ea_proj<OUTD><<<4, OUTD, 0, stream>>>(flow_emb, posvec, cnt, c2_We, eaW2, 1.f / (float)E);

    // --- layer 1 ----------------------------------------------------------
    gemm_dual_wmma<<<N / 16, 256, 0, stream>>>(h, c1_Wl, c1_bl, xl, c1_Wr, c1_br, xr,
                                               NODE_DIM, H1);
    fill_f32<<<256, 256, 0, stream>>>(emax, -3.0e38f, N);
    fill_f32<<<256, 256, 0, stream>>>(denom, 0.f, N);
    edge_scores<H1><<<eBlocks, 256, 0, stream>>>(eidx, eattr, xl, xr, eaW1, c1_att, esc, emax, E);
    self_scores<H1><<<nBlocks, 256, 0, stream>>>(xl, xr, eaW1, c1_att, eself, emax, N);
    exp_accum<<<1024, 256, 0, stream>>>(eidx + E, esc, emax, denom, E);
    exp_accum<<<1024, 256, 0, stream>>>(nullptr, eself, emax, denom, N);
    {
        long tot = (long)N * H1;
        hout_init<H1><<<(int)((tot + 255) / 256), 256, 0, stream>>>(xl, eself, denom, c1_bias, h1, N);
    }
    edge_aggregate<H1><<<eBlocks, 256, 0, stream>>>(eidx, esc, denom, xl, h1, E);

    // --- layer 2 (xl/xr buffers reused at N*OUTD) -------------------------
    gemm_dual_wmma<<<N / 16, 256, 0, stream>>>(h1, c2_Wl, c2_bl, xl, c2_Wr, c2_br, xr,
                                               H1, OUTD);
    fill_f32<<<256, 256, 0, stream>>>(emax, -3.0e38f, N);
    fill_f32<<<256, 256, 0, stream>>>(denom, 0.f, N);
    edge_scores<OUTD><<<eBlocks, 256, 0, stream>>>(eidx, eattr, xl, xr, eaW2, c2_att, esc, emax, E);
    self_scores<OUTD><<<nBlocks, 256, 0, stream>>>(xl, xr, eaW2, c2_att, eself, emax, N);
    exp_accum<<<1024, 256, 0, stream>>>(eidx + E, esc, emax, denom, E);
    exp_accum<<<1024, 256, 0, stream>>>(nullptr, eself, emax, denom, N);
    {
        long tot = (long)N * OUTD;
        hout_init<OUTD><<<(int)((tot + 255) / 256), 256, 0, stream>>>(xl, eself, denom, c2_bias, h2, N);
    }
    edge_aggregate<OUTD><<<eBlocks, 256, 0, stream>>>(eidx, esc, denom, xl, h2, E);

    // --- pool + policy head ----------------------------------------------
    fill_f32<<<1, ACTION_DIM, 0, stream>>>(gsum, 0.f, ACTION_DIM);
    reduce_mean<<<512, 256, 0, stream>>>(h2, gsum, N);
    policy_softmax<<<1, ACTION_DIM, 0, stream>>>(gsum, policy_W, policy_b, (float*)d_out,
                                                 1.f / (float)N);
    (void)n_in; (void)out_size; (void)ws_size;
}